// LinkGenerator_68771016343680
// MI455X (gfx1250) — compile-verified
//
#include <hip/hip_runtime.h>
#include <hip/hip_bf16.h>

typedef __attribute__((ext_vector_type(2))) float v2f;
typedef __attribute__((ext_vector_type(8))) float v8f;

#define HD 128           // feature dim
#define SLOPE 0.01f

__device__ __forceinline__ float leaky(float v) { return v >= 0.0f ? v : SLOPE * v; }

// ---------------------------------------------------------------------------
// Elementwise helpers
// ---------------------------------------------------------------------------
__global__ void fill_f32(float* __restrict__ p, float v, int n) {
    int i = blockIdx.x * blockDim.x + threadIdx.x;
    if (i < n) p[i] = v;
}

__global__ void deg_count(const int* __restrict__ dst, float* __restrict__ deg, int E) {
    int e = blockIdx.x * blockDim.x + threadIdx.x;
    if (e < E) atomicAdd(&deg[dst[e]], 1.0f);
}

__global__ void to_dinv(float* __restrict__ d, int n) {
    int i = blockIdx.x * blockDim.x + threadIdx.x;
    if (i < n) d[i] = 1.0f / sqrtf(d[i]);
}

// out = 0.5*a + 0.5*b
__global__ void blend_half(const float* __restrict__ a, const float* __restrict__ b,
                           float* __restrict__ o, int n) {
    int i = blockIdx.x * blockDim.x + threadIdx.x;
    if (i < n) o[i] = 0.5f * a[i] + 0.5f * b[i];
}

// ---------------------------------------------------------------------------
// Edge aggregation: agg[dst] += h[src] * dinv[src]*dinv[dst]
// one wave (32 lanes) per edge; each lane moves 4 floats (128 total)
// ---------------------------------------------------------------------------
__global__ __launch_bounds__(256)
void edge_scatter(const float* __restrict__ h, const int* __restrict__ src,
                  const int* __restrict__ dst, const float* __restrict__ dinv,
                  float* __restrict__ agg, int E) {
    int e    = blockIdx.x * 8 + (threadIdx.x >> 5);
    int lane = threadIdx.x & 31;
    if (e >= E) return;
    int s = src[e], d = dst[e];
    float c = dinv[s] * dinv[d];
    float4 v = ((const float4*)(h + (size_t)s * HD))[lane];
    float* ap = agg + (size_t)d * HD + lane * 4;
    atomicAdd(ap + 0, v.x * c);
    atomicAdd(ap + 1, v.y * c);
    atomicAdd(ap + 2, v.z * c);
    atomicAdd(ap + 3, v.w * c);
}

// ---------------------------------------------------------------------------
// GCN epilogue: out = leaky(agg + hw*dinv^2 + bias [+ res])
// safe for out == agg (read-before-write per element)
// ---------------------------------------------------------------------------
__global__ void gcn_finish(const float* __restrict__ agg, const float* __restrict__ hw,
                           const float* __restrict__ dinv, const float* __restrict__ bias,
                           const float* __restrict__ res, float* __restrict__ out,
                           int total, int useRes) {
    int i = blockIdx.x * blockDim.x + threadIdx.x;
    if (i >= total) return;
    int node = i >> 7;
    int f    = i & (HD - 1);
    float di = dinv[node];
    float v  = agg[i] + hw[i] * di * di + bias[f];
    if (useRes) v += res[i];
    out[i] = leaky(v);
}

// ---------------------------------------------------------------------------
// WMMA GEMM: out[M x 128] = A[M x 128] @ W[128 x 128], fp32 accumulate.
// mode 0: out = acc
// mode 1: out = leaky(acc + bias)
// mode 2: out = leaky(acc + bias + res)
// Block = 256 threads (8 waves). Each wave computes a 16-row x 128-col slab.
// W staged through 64KB LDS.
//
// A fragments for the whole K=128 are preloaded branch-free into 64 VGPRs
// (out-of-range rows clamp to row 0; their D rows are never stored, and a
// D row depends only on the matching A row, so no masking is needed).
//
// Fragment layouts per CDNA5 ISA 7.12.2 (V_WMMA_F32_16X16X4_F32):
//   A (16x4): lane m=lane&15; a[r] = A[m][k0 + r + 2*(lane>>4)]
//   B (4x16): lane n=lane&15; b[r] = W[k0 + r + 2*(lane>>4)][n]
//   C/D (16x16): acc[r] = D[r + 8*(lane>>4)][lane&15 + 16*nt]
// ---------------------------------------------------------------------------
__global__ __launch_bounds__(256)
void gemm_ep(const float* __restrict__ A, const float* __restrict__ W,
             const float* __restrict__ bias, const float* __restrict__ res,
             float* __restrict__ out, int M, int mode) {
    __shared__ float Wl[HD * HD];   // 64 KB
    {
        const float4* Wv = (const float4*)W;
        float4*       Lv = (float4*)Wl;
        for (int i = threadIdx.x; i < HD * HD / 4; i += 256) Lv[i] = Wv[i];
    }
    __syncthreads();

    const int wave = threadIdx.x >> 5;
    const int lane = threadIdx.x & 31;
    const int hi   = lane >> 4;        // 0 or 1
    const int lm   = lane & 15;
    const int row0 = blockIdx.x * 128 + wave * 16;

    // ---- branch-free A preload (32 x v2f = 64 VGPRs), single load clause
    const int arow = row0 + lm;
    const float* Arow = A + (size_t)(arow < M ? arow : 0) * HD + 2 * hi;
    v2f af[32];
#pragma unroll
    for (int j = 0; j < 32; ++j) {
        float2 t = *(const float2*)(Arow + 4 * j);
        af[j][0] = t.x;
        af[j][1] = t.y;
    }

    v8f acc[8];
#pragma unroll
    for (int t = 0; t < 8; ++t) {
        v8f z = {0.f, 0.f, 0.f, 0.f, 0.f, 0.f, 0.f, 0.f};
        acc[t] = z;
    }

#pragma unroll
    for (int ks = 0; ks < 32; ++ks) {
        const int kk = ks * 4 + 2 * hi;
#pragma unroll
        for (int nt = 0; nt < 8; ++nt) {
            const int n = lm + nt * 16;
            v2f b;
            b[0] = Wl[kk * HD + n];
            b[1] = Wl[(kk + 1) * HD + n];
            acc[nt] = __builtin_amdgcn_wmma_f32_16x16x4_f32(
                /*neg_a=*/false, af[ks], /*neg_b=*/false, b,
                /*c_mod=*/(short)0, acc[nt], /*reuse_a=*/false, /*reuse_b=*/false);
        }
    }

    // epilogue
#pragma unroll
    for (int r = 0; r < 8; ++r) {
        const int row = row0 + r + 8 * hi;
        if (row < M) {
#pragma unroll
            for (int nt = 0; nt < 8; ++nt) {
                const int n = lm + nt * 16;
                float v = acc[nt][r];
                if (mode >= 1) v += bias[n];
                if (mode == 2) v += res[(size_t)row * HD + n];
                if (mode >= 1) v = leaky(v);
                out[(size_t)row * HD + n] = v;
            }
        }
    }
}

// ---------------------------------------------------------------------------
// Driver
// ---------------------------------------------------------------------------
extern "C" void kernel_launch(void* const* d_in, const int* in_sizes, int n_in,
                              void* d_out, int out_size, void* d_ws, size_t ws_size,
                              hipStream_t stream) {
    const float* x    = (const float*)d_in[0];
    const int*   edge = (const int*)  d_in[1];
    const float* gen  = (const float*)d_in[2];
    const float* Wc1  = (const float*)d_in[3];  const float* bc1 = (const float*)d_in[4];
    const float* Wc2  = (const float*)d_in[5];  const float* bc2 = (const float*)d_in[6];
    const float* Wm1  = (const float*)d_in[7];  const float* bm1 = (const float*)d_in[8];
    const float* Wm2  = (const float*)d_in[9];  const float* bm2 = (const float*)d_in[10];
    const float* Wp1  = (const float*)d_in[11]; const float* bp1 = (const float*)d_in[12];
    const float* Wp2  = (const float*)d_in[13]; const float* bp2 = (const float*)d_in[14];
    float* out = (float*)d_out;

    const int N = in_sizes[0] / HD;
    const int E = in_sizes[1] / 2;
    const int* src = edge;
    const int* dst = edge + E;

    // workspace layout
    char* ws = (char*)d_ws;
    float* dinv = (float*)ws;
    size_t off  = ((size_t)N * 4 + 255) & ~(size_t)255;
    const size_t fsz = (size_t)N * HD * sizeof(float);
    float* bufA = (float*)(ws + off);
    float* bufB = (float*)(ws + off + fsz);
    float* bufC = (float*)(ws + off + 2 * fsz);

    const int total = N * HD;
    const dim3 blk(256);
    const dim3 gN((N + 255) / 256);
    const dim3 gE((E + 255) / 256);
    const dim3 gT((total + 255) / 256);
    const dim3 gGemm((N + 127) / 128);
    const dim3 gEdge((E + 7) / 8);

    // degrees -> dinv
    fill_f32<<<gN, blk, 0, stream>>>(dinv, 1.0f, N);
    deg_count<<<gE, blk, 0, stream>>>(dst, dinv, E);
    to_dinv<<<gN, blk, 0, stream>>>(dinv, N);

    // ---- GCN layer 1: h1 = leaky(agg(x@Wc1) + bc1)
    gemm_ep<<<gGemm, blk, 0, stream>>>(x, Wc1, nullptr, nullptr, bufA, N, 0);   // hw1 -> A
    fill_f32<<<gT, blk, 0, stream>>>(bufB, 0.0f, total);
    edge_scatter<<<gEdge, blk, 0, stream>>>(bufA, src, dst, dinv, bufB, E);
    gcn_finish<<<gT, blk, 0, stream>>>(bufB, bufA, dinv, bc1, nullptr, bufB, total, 0); // h1 -> B

    // ---- GCN layer 2 + residual: h2 = leaky(agg(h1@Wc2) + bc2 + h1)
    gemm_ep<<<gGemm, blk, 0, stream>>>(bufB, Wc2, nullptr, nullptr, bufA, N, 0); // hw2 -> A
    fill_f32<<<gT, blk, 0, stream>>>(bufC, 0.0f, total);
    edge_scatter<<<gEdge, blk, 0, stream>>>(bufA, src, dst, dinv, bufC, E);
    gcn_finish<<<gT, blk, 0, stream>>>(bufC, bufA, dinv, bc2, bufB, bufC, total, 1);   // h2 -> C

    // ---- MLP block with residual: h3 = leaky(leaky(h2@Wm1+bm1)@Wm2 + bm2 + h2)
    gemm_ep<<<gGemm, blk, 0, stream>>>(bufC, Wm1, bm1, nullptr, bufA, N, 1);     // t  -> A
    gemm_ep<<<gGemm, blk, 0, stream>>>(bufA, Wm2, bm2, bufC, bufB, N, 2);        // h3 -> B

    // ---- lambda blend with gen_data
    blend_half<<<gT, blk, 0, stream>>>(bufB, gen, bufA, total);                  // mix -> A

    // ---- proj MLP: out = leaky(leaky(mix@Wp1+bp1)@Wp2+bp2)
    gemm_ep<<<gGemm, blk, 0, stream>>>(bufA, Wp1, bp1, nullptr, bufC, N, 1);     // p -> C
    gemm_ep<<<gGemm, blk, 0, stream>>>(bufC, Wp2, bp2, nullptr, out, N, 1);      // final -> d_out
}